// SideWindowFilter_61014305407465
// MI455X (gfx1250) — compile-verified
//
#include <hip/hip_runtime.h>
#include <cstdint>
#include <cstddef>

// ---------------------------------------------------------------------------
// Side-window filter, MI455X (gfx1250).
// 8 iteration kernels (L2-resident ping-pong, 192MB L2 holds both buffers)
// + 1 vectorized diff/clip kernel (the only HBM-bound pass).
// Each 5x5 directional box conv is rebuilt from separable partial sums.
// Tile loads use CDNA5 async global->LDS (ASYNCcnt) instructions.
// ---------------------------------------------------------------------------

#define IMG_W 2048
#define IMG_H 2048
#define IMG_C 3
#define IMG_HW (IMG_W * IMG_H)

#define TILE_X 64
#define TILE_Y 16
#define HALO 2
#define LOAD_W (TILE_X + 2 * HALO)   // 68
#define LOAD_H (TILE_Y + 2 * HALO)   // 20
#define SP 72                        // padded LDS row stride (floats)
#define NTHREADS 256

// Low 32 bits of a flat pointer into __shared__ are the LDS byte offset on
// gfx1250 (shared aperture occupies flat-addr bits [63:32]).
__device__ __forceinline__ unsigned lds_off_of(const void* p) {
  return (unsigned)(uintptr_t)p;
}

// CDNA5 async tensor-path load: global -> LDS, tracked by ASYNCcnt.
__device__ __forceinline__ void async_ld_f32(unsigned lds_byte_off, const float* g) {
  asm volatile("global_load_async_to_lds_b32 %0, %1, off"
               :: "v"(lds_byte_off), "v"(g)
               : "memory");
}

__device__ __forceinline__ void wait_async0() {
#if __has_builtin(__builtin_amdgcn_s_wait_asynccnt)
  __builtin_amdgcn_s_wait_asynccnt(0);
#else
  asm volatile("s_wait_asynccnt 0" ::: "memory");
#endif
}

// One side-window iteration over one channel plane.
// SRC_HWC: first iteration reads the original HWC image; later iterations
// read the CHW ping-pong buffer. Output is always CHW.
template <bool SRC_HWC>
__global__ __launch_bounds__(NTHREADS)
void swf_iter_kernel(const float* __restrict__ src, float* __restrict__ dst) {
  __shared__ float s_img[LOAD_H][SP];        // raw tile (replicate-clamped)
  __shared__ float s_hl[LOAD_H][TILE_X];     // 3-wide left row sums
  __shared__ float s_hr[LOAD_H][TILE_X];     // 3-wide right row sums

  const int tid = threadIdx.x;
  const int tx0 = blockIdx.x * TILE_X;
  const int ty0 = blockIdx.y * TILE_Y;
  const int ch  = blockIdx.z;

  // ---- async global -> LDS tile load with replicate (edge) clamping ----
  for (int i = tid; i < LOAD_H * LOAD_W; i += NTHREADS) {
    const int ly = i / LOAD_W;
    const int lx = i - ly * LOAD_W;
    int gy = ty0 - HALO + ly;
    int gx = tx0 - HALO + lx;
    gy = gy < 0 ? 0 : (gy > IMG_H - 1 ? IMG_H - 1 : gy);
    gx = gx < 0 ? 0 : (gx > IMG_W - 1 ? IMG_W - 1 : gx);
    const float* g = SRC_HWC
        ? (src + ((size_t)gy * IMG_W + gx) * IMG_C + ch)
        : (src + (size_t)ch * IMG_HW + (size_t)gy * IMG_W + gx);
    async_ld_f32(lds_off_of(&s_img[ly][lx]), g);
  }
  wait_async0();
  __syncthreads();

  // ---- horizontal partial sums (shared across the 8 directions) ----
  // hl[y][x] = img cols (x+2)-2 .. (x+2)   (left 3-window at output col x)
  // hr[y][x] = img cols (x+2)   .. (x+2)+2 (right 3-window)
  for (int i = tid; i < LOAD_H * TILE_X; i += NTHREADS) {
    const int y = i >> 6;
    const int x = i & 63;
    const float* r = &s_img[y][x];
    s_hl[y][x] = r[0] + r[1] + r[2];
    s_hr[y][x] = r[2] + r[3] + r[4];
  }
  __syncthreads();

  const float inv15 = 1.0f / 15.0f;
  const float inv9  = 1.0f / 9.0f;

  // ---- each thread: one column x, 4 vertically adjacent output rows ----
  // Register-cache 8 rows of hl/hr/center-column: 24 DS reads per 4 pixels,
  // 3-row sliding sums shared between the 4 outputs.
  {
    const int x  = tid & 63;              // output column in tile
    const int y0 = (tid >> 6) * 4;        // first of 4 output rows (0,4,8,12)

    float hl[8], hr[8], cc[8];
#pragma unroll
    for (int k = 0; k < 8; ++k) {
      hl[k] = s_hl[y0 + k][x];
      hr[k] = s_hr[y0 + k][x];
      cc[k] = s_img[y0 + k][x + 2];
    }

    float sl[6], sr[6], sc[6];            // 3-row sums starting at row y0+k
#pragma unroll
    for (int k = 0; k < 6; ++k) {
      sl[k] = hl[k] + hl[k + 1] + hl[k + 2];
      sr[k] = hr[k] + hr[k + 1] + hr[k + 2];
      sc[k] = cc[k] + cc[k + 1] + cc[k + 2];
    }

    float* drow = dst + (size_t)ch * IMG_HW + (size_t)(ty0 + y0) * IMG_W + (tx0 + x);
#pragma unroll
    for (int j = 0; j < 4; ++j) {
      const float tl = sl[j];             // 3x3 NW box
      const float bl = sl[j + 2];         // 3x3 SW box
      const float tr = sr[j];             // 3x3 NE box
      const float br = sr[j + 2];         // 3x3 SE box
      const float vtc = sc[j];
      const float vbc = sc[j + 2];
      const float cl = hl[j + 2];
      const float cr = hr[j + 2];
      const float im = cc[j + 2];

      float conv[8];
      conv[0] = (tl + bl - cl) * inv15;   // L : 5 rows x left 3 cols
      conv[1] = (tr + br - cr) * inv15;   // R : 5 rows x right 3 cols
      conv[2] = (tl + tr - vtc) * inv15;  // U : top 3 rows x 5 cols
      conv[3] = (bl + br - vbc) * inv15;  // D : bottom 3 rows x 5 cols
      conv[4] = tl * inv9;                // NW
      conv[5] = tr * inv9;                // NE
      conv[6] = bl * inv9;                // SW
      conv[7] = br * inv9;                // SE

      float best = conv[0] - im;
      float ba   = fabsf(best);
#pragma unroll
      for (int k = 1; k < 8; ++k) {
        const float d = conv[k] - im;
        const float a = fabsf(d);
        if (a < ba) { ba = a; best = d; } // strict '<' keeps first-min (argmin)
      }
      drow[(size_t)j * IMG_W] = im + best;
    }
  }
}

// out[h][w][c] = clip(|img_hwc - res_chw|, 0, 255), 4 pixels per thread,
// pure b128 traffic on image, result planes, and output.
__global__ __launch_bounds__(NTHREADS)
void swf_final_kernel(const float4* __restrict__ img4,
                      const float4* __restrict__ r0,
                      const float4* __restrict__ r1,
                      const float4* __restrict__ r2,
                      float4* __restrict__ out4) {
  const int t = blockIdx.x * blockDim.x + threadIdx.x;
  if (t >= IMG_HW / 4) return;

  const float4 a0 = img4[(size_t)t * 3 + 0];
  const float4 a1 = img4[(size_t)t * 3 + 1];
  const float4 a2 = img4[(size_t)t * 3 + 2];
  const float4 b0 = r0[t];
  const float4 b1 = r1[t];
  const float4 b2 = r2[t];

  const float v[12]   = {a0.x, a0.y, a0.z, a0.w, a1.x, a1.y,
                         a1.z, a1.w, a2.x, a2.y, a2.z, a2.w};
  const float bv[3][4] = {{b0.x, b0.y, b0.z, b0.w},
                          {b1.x, b1.y, b1.z, b1.w},
                          {b2.x, b2.y, b2.z, b2.w}};
  float o[12];
#pragma unroll
  for (int m = 0; m < 4; ++m) {
#pragma unroll
    for (int c = 0; c < 3; ++c) {
      float d = fabsf(v[3 * m + c] - bv[c][m]);
      o[3 * m + c] = d > 255.0f ? 255.0f : d;
    }
  }
  out4[(size_t)t * 3 + 0] = make_float4(o[0], o[1], o[2], o[3]);
  out4[(size_t)t * 3 + 1] = make_float4(o[4], o[5], o[6], o[7]);
  out4[(size_t)t * 3 + 2] = make_float4(o[8], o[9], o[10], o[11]);
}

extern "C" void kernel_launch(void* const* d_in, const int* in_sizes, int n_in,
                              void* d_out, int out_size, void* d_ws, size_t ws_size,
                              hipStream_t stream) {
  (void)in_sizes; (void)n_in; (void)out_size; (void)ws_size;

  const float* img = (const float*)d_in[0];     // HWC 2048x2048x3 f32

  // Ping-pong CHW plane buffers. d_out doubles as scratch mid-pipeline and is
  // fully overwritten by the final pass. d_ws needs 3*2048*2048*4 = 48 MiB.
  float* A = (float*)d_out;
  float* B = (float*)d_ws;

  dim3 grid(IMG_W / TILE_X, IMG_H / TILE_Y, IMG_C);
  dim3 blk(NTHREADS);

  // 8 side-window iterations: img -> A -> B -> A -> B -> A -> B -> A -> B
  swf_iter_kernel<true ><<<grid, blk, 0, stream>>>(img, A);  // iter 1
  swf_iter_kernel<false><<<grid, blk, 0, stream>>>(A, B);    // iter 2
  swf_iter_kernel<false><<<grid, blk, 0, stream>>>(B, A);    // iter 3
  swf_iter_kernel<false><<<grid, blk, 0, stream>>>(A, B);    // iter 4
  swf_iter_kernel<false><<<grid, blk, 0, stream>>>(B, A);    // iter 5
  swf_iter_kernel<false><<<grid, blk, 0, stream>>>(A, B);    // iter 6
  swf_iter_kernel<false><<<grid, blk, 0, stream>>>(B, A);    // iter 7
  swf_iter_kernel<false><<<grid, blk, 0, stream>>>(A, B);    // iter 8 -> res in d_ws

  swf_final_kernel<<<(IMG_HW / 4 + NTHREADS - 1) / NTHREADS, blk, 0, stream>>>(
      (const float4*)img, (const float4*)(B + 0 * (size_t)IMG_HW),
      (const float4*)(B + 1 * (size_t)IMG_HW),
      (const float4*)(B + 2 * (size_t)IMG_HW), (float4*)d_out);
}